// RandomShiftsAug_66718021976207
// MI455X (gfx1250) — compile-verified
//
#include <hip/hip_runtime.h>
#include <cstdint>

// RandomShiftsAug on MI455X (gfx1250).
// Memory-bound gather: 260 MB total traffic -> ~11 us at 23.3 TB/s.
// Strategy: per (n,c) plane, async-DMA the whole 84x84 fp32 tile into LDS
// (global_load_async_to_lds_b128, tracked by ASYNCcnt), then serve all four
// bilinear taps from LDS and write coalesced b128 stores. One HBM read and
// one HBM write per element = bandwidth floor.

#define N_IMG   512
#define C_CH    9
#define H_DIM   84
#define W_DIM   84
#define PAD     4
#define HP      (H_DIM + 2 * PAD)      // 92 (== Wp)
#define PLANE   (H_DIM * W_DIM)        // 7056 floats
#define PLANE_V (PLANE / 4)            // 1764 float4 (exact)
#define BLOCK   256

__device__ __forceinline__ float sample_lds(const float* s, int y, int x) {
    // Reference semantics: padding_mode='zeros' on the 92x92 padded frame,
    // where padded frame itself is replicate-padded from the 84x84 plane.
    bool valid = (y >= 0) & (y <= HP - 1) & (x >= 0) & (x <= HP - 1);
    int yc = min(max(y, 0), HP - 1);
    int xc = min(max(x, 0), HP - 1);
    int yi = min(max(yc - PAD, 0), H_DIM - 1);   // replicate pad -> clamp
    int xi = min(max(xc - PAD, 0), W_DIM - 1);
    float v = s[yi * W_DIM + xi];
    return valid ? v : 0.0f;
}

__global__ __launch_bounds__(BLOCK)
void rsa_kernel(const float* __restrict__ x,
                const int*   __restrict__ shift,
                float*       __restrict__ out) {
    __shared__ float smem[PLANE];

    const int plane = blockIdx.x;            // n*C + c
    const int n     = plane / C_CH;
    const int tid   = threadIdx.x;

    // ---------------- async stage: plane -> LDS ----------------------
    const uint64_t gbase = (uint64_t)(uintptr_t)(x) + (uint64_t)plane * (PLANE * 4u);
    const uint32_t lbase = (uint32_t)(uintptr_t)(&smem[0]);

    for (int idx = tid; idx < PLANE_V; idx += BLOCK) {
        uint32_t laddr = lbase + (uint32_t)idx * 16u;
        uint64_t gaddr = gbase + (uint64_t)idx * 16u;
        asm volatile("global_load_async_to_lds_b128 %0, %1, off"
                     :: "v"(laddr), "v"(gaddr)
                     : "memory");
    }
    // Wait for this wave's async DMA into LDS, then cross-wave barrier.
    asm volatile("s_wait_asynccnt 0" ::: "memory");
    __syncthreads();

    // ---------------- per-image shift (uniform scalars) --------------
    const float sxf = (float)shift[2 * n + 0] * (2.0f / (float)HP);
    const float syf = (float)shift[2 * n + 1] * (2.0f / (float)HP);

    // linspace(-1+eps, 1-eps, Hp) as in the reference
    const float eps  = 1.0f / (float)HP;
    const float step = (2.0f - 2.0f * eps) / (float)(HP - 1);
    const float a0   = -1.0f + eps;

    float* __restrict__ dst = out + (size_t)plane * PLANE;

    // ---------------- bilinear from LDS, b128 stores -----------------
    for (int vidx = tid; vidx < PLANE_V; vidx += BLOCK) {
        float r[4];
#pragma unroll
        for (int k = 0; k < 4; ++k) {
            int e = vidx * 4 + k;
            int i = e / W_DIM;
            int j = e - i * W_DIM;

            float gx = a0 + step * (float)j + sxf;
            float gy = a0 + step * (float)i + syf;
            float ix = ((gx + 1.0f) * (float)HP - 1.0f) * 0.5f;   // Wp == Hp
            float iy = ((gy + 1.0f) * (float)HP - 1.0f) * 0.5f;

            float x0f = floorf(ix), y0f = floorf(iy);
            float wx1 = ix - x0f,   wx0 = 1.0f - wx1;
            float wy1 = iy - y0f,   wy0 = 1.0f - wy1;
            int   x0  = (int)x0f,   y0  = (int)y0f;
            int   x1  = x0 + 1,     y1  = y0 + 1;

            float v00 = sample_lds(smem, y0, x0);
            float v01 = sample_lds(smem, y0, x1);
            float v10 = sample_lds(smem, y1, x0);
            float v11 = sample_lds(smem, y1, x1);

            r[k] = v00 * (wy0 * wx0) + v01 * (wy0 * wx1)
                 + v10 * (wy1 * wx0) + v11 * (wy1 * wx1);
        }
        float4 vout = make_float4(r[0], r[1], r[2], r[3]);
        *reinterpret_cast<float4*>(dst + (size_t)vidx * 4) = vout;
    }
}

extern "C" void kernel_launch(void* const* d_in, const int* in_sizes, int n_in,
                              void* d_out, int out_size, void* d_ws, size_t ws_size,
                              hipStream_t stream) {
    (void)in_sizes; (void)n_in; (void)out_size; (void)d_ws; (void)ws_size;
    const float* x     = (const float*)d_in[0];
    const int*   shift = (const int*)d_in[1];
    // d_in[2] is pad == 4, baked into the kernel constants.
    float* out = (float*)d_out;

    dim3 grid(N_IMG * C_CH);   // 4608 planes
    dim3 block(BLOCK);
    rsa_kernel<<<grid, block, 0, stream>>>(x, shift, out);
}